// RouteOptimizerGCN_19189913879362
// MI455X (gfx1250) — compile-verified
//
#include <hip/hip_runtime.h>

typedef float v2f __attribute__((ext_vector_type(2)));
typedef float v8f __attribute__((ext_vector_type(8)));

#define HID 32

// ---------------- utility: fill buffer with constant ----------------
__global__ void k_fill(float* __restrict__ p, float v, long long n) {
    long long i = (long long)blockIdx.x * blockDim.x + threadIdx.x;
    if (i < n) p[i] = v;
}

// ---------------- degree count: deg[dst] += 1 per edge ----------------
__global__ void k_deg(const int* __restrict__ dst, float* __restrict__ deg, long long e) {
    long long i = (long long)blockIdx.x * blockDim.x + threadIdx.x;
    if (i < e) atomicAdd(&deg[dst[i]], 1.0f);
}

// ---------------- dinv = deg^-1/2 (deg >= 1 always due to self loop) ----------------
__global__ void k_rsqrt(float* __restrict__ d, long long n) {
    long long i = (long long)blockIdx.x * blockDim.x + threadIdx.x;
    if (i < n) {
        float v = d[i];
        d[i] = (v > 0.0f) ? rsqrtf(v) : 0.0f;
    }
}

// ---------------- WMMA fp32 GEMM: H[n,32] = A[n,KDIM] x W[KDIM,32] ----------------
// One wave computes a 16x32 tile using V_WMMA_F32_16X16X4_F32, K-step = 4.
// Block = 128 threads (4 waves) -> 64 rows per block.
template <int KDIM>
__global__ __launch_bounds__(128) void k_gemm(const float* __restrict__ A,
                                              const float* __restrict__ W,
                                              float* __restrict__ H, int n) {
    const int lane = threadIdx.x & 31;
    const int wave = threadIdx.x >> 5;
    const int row0 = (blockIdx.x * 4 + wave) * 16;
    if (row0 >= n) return;  // wave-uniform exit: EXEC stays all-ones for WMMA

    const int lr = lane & 15;          // M (for A) / N (for B,C,D) within tile
    const int kh = (lane >> 4) << 1;   // K sub-offset: lanes 0-15 -> {0,1}, 16-31 -> {2,3}

    int arow = row0 + lr;
    if (arow >= n) arow = n - 1;       // clamp (n % 16 == 0 in practice)
    const float* __restrict__ ap = A + (size_t)arow * KDIM;

    v8f c0 = {0.f, 0.f, 0.f, 0.f, 0.f, 0.f, 0.f, 0.f};
    v8f c1 = {0.f, 0.f, 0.f, 0.f, 0.f, 0.f, 0.f, 0.f};

#pragma unroll
    for (int k0 = 0; k0 < KDIM; k0 += 4) {
        // A fragment 16x4: VGPR0 = K (kh), VGPR1 = K (kh+1)
        v2f a;
        a.x = ap[k0 + kh];
        a.y = ap[k0 + kh + 1];
        // B fragment 4x16: row K striped across lanes, N = lane&15
        const float* __restrict__ wp = W + (size_t)(k0 + kh) * HID;
        v2f b0, b1;
        b0.x = wp[lr];            // K=k0+kh,   N = lr      (cols 0..15)
        b0.y = wp[HID + lr];      // K=k0+kh+1, N = lr
        b1.x = wp[16 + lr];       // cols 16..31
        b1.y = wp[HID + 16 + lr];
        c0 = __builtin_amdgcn_wmma_f32_16x16x4_f32(false, a, false, b0, (short)0, c0, false, false);
        c1 = __builtin_amdgcn_wmma_f32_16x16x4_f32(false, a, false, b1, (short)0, c1, false, false);
    }

    // C/D layout: VGPR r -> M = r (lanes 0-15) or r+8 (lanes 16-31); N = lane&15
    const int mofs = (lane >> 4) * 8;
#pragma unroll
    for (int r = 0; r < 8; ++r) {
        int row = row0 + mofs + r;
        if (row < n) {
            float* __restrict__ hp = H + (size_t)row * HID;
            hp[lr]      = c0[r];
            hp[16 + lr] = c1[r];
        }
    }
}

// ---------------- edge scatter: agg[dst] += h[src] * dinv[src]*dinv[dst] ----------------
// 8 lanes cooperate on one edge: each lane moves one float4 chunk (coalesced 128B gather).
__global__ __launch_bounds__(256) void k_scatter(const int* __restrict__ src,
                                                 const int* __restrict__ dst,
                                                 const float* __restrict__ dinv,
                                                 const float* __restrict__ h,
                                                 float* __restrict__ agg, long long e) {
    long long t = (long long)blockIdx.x * blockDim.x + threadIdx.x;
    long long idx = t >> 3;
    if (idx >= e) return;
    const int c = (int)(t & 7);
    const int s = src[idx];
    const int d = dst[idx];
    const float nrm = dinv[s] * dinv[d];
    const float4 v = ((const float4*)(h + (size_t)s * HID))[c];
    float* __restrict__ ap = agg + (size_t)d * HID + c * 4;
    atomicAdd(ap + 0, v.x * nrm);
    atomicAdd(ap + 1, v.y * nrm);
    atomicAdd(ap + 2, v.z * nrm);
    atomicAdd(ap + 3, v.w * nrm);
}

// ---------------- finalize: agg = relu(agg + h*dinv^2 + b) (fused self-loop) ----------------
__global__ void k_finalize(const float* __restrict__ h, const float* __restrict__ dinv,
                           const float* __restrict__ b, float* __restrict__ agg,
                           long long total) {
    long long t = (long long)blockIdx.x * blockDim.x + threadIdx.x;
    if (t >= total) return;
    long long i = t >> 5;      // node
    int k = (int)(t & 31);     // feature
    float di = dinv[i];
    float v = agg[t] + h[t] * (di * di) + b[k];
    agg[t] = v > 0.0f ? v : 0.0f;
}

// ---------------- final FC: out[i] = dot(h[i,:], Wfc) + bfc ----------------
__global__ void k_fc(const float* __restrict__ h, const float* __restrict__ wfc,
                     const float* __restrict__ bfc, float* __restrict__ out, int n) {
    int i = blockIdx.x * blockDim.x + threadIdx.x;
    if (i >= n) return;
    const float4* __restrict__ hp = (const float4*)(h + (size_t)i * HID);
    const float4* __restrict__ wp = (const float4*)wfc;
    float acc = 0.0f;
#pragma unroll
    for (int k = 0; k < 8; ++k) {
        float4 a = hp[k], w = wp[k];
        acc += a.x * w.x + a.y * w.y + a.z * w.z + a.w * w.w;
    }
    out[i] = acc + bfc[0];
}

extern "C" void kernel_launch(void* const* d_in, const int* in_sizes, int n_in,
                              void* d_out, int out_size, void* d_ws, size_t ws_size,
                              hipStream_t stream) {
    const float* x   = (const float*)d_in[0];
    const int*   ei  = (const int*)d_in[1];
    const float* W1  = (const float*)d_in[2];
    const float* b1  = (const float*)d_in[3];
    const float* W2  = (const float*)d_in[4];
    const float* b2  = (const float*)d_in[5];
    const float* Wfc = (const float*)d_in[6];
    const float* bfc = (const float*)d_in[7];
    float* out = (float*)d_out;

    const int F = 128;
    const long long n = in_sizes[0] / F;       // 200000
    const long long e = in_sizes[1] / 2;       // 6.4M
    const int* src = ei;
    const int* dst = ei + e;

    // workspace layout: dinv[n] | h[n*32] | agg[n*32]
    float* dinv = (float*)d_ws;
    float* hbuf = dinv + n;
    float* agg  = hbuf + n * HID;

    const long long nh = n * HID;
    const int B = 256;
    dim3 bfill((unsigned)((n + B - 1) / B));
    dim3 bfillh((unsigned)((nh + B - 1) / B));
    dim3 bedge((unsigned)((e + B - 1) / B));
    dim3 bscat((unsigned)((e * 8 + B - 1) / B));
    dim3 bgemm((unsigned)((n + 63) / 64));

    // degrees -> dinv
    k_fill<<<bfill, B, 0, stream>>>(dinv, 1.0f, n);            // self loop
    k_deg<<<bedge, B, 0, stream>>>(dst, dinv, e);
    k_rsqrt<<<bfill, B, 0, stream>>>(dinv, n);

    // layer 1: h = x @ W1 ; aggregate ; relu(+b1)
    k_gemm<128><<<bgemm, 128, 0, stream>>>(x, W1, hbuf, (int)n);
    k_fill<<<bfillh, B, 0, stream>>>(agg, 0.0f, nh);
    k_scatter<<<bscat, B, 0, stream>>>(src, dst, dinv, hbuf, agg, e);
    k_finalize<<<bfillh, B, 0, stream>>>(hbuf, dinv, b1, agg, nh);

    // layer 2: h = agg @ W2 ; aggregate ; relu(+b2)
    k_gemm<32><<<bgemm, 128, 0, stream>>>(agg, W2, hbuf, (int)n);
    k_fill<<<bfillh, B, 0, stream>>>(agg, 0.0f, nh);
    k_scatter<<<bscat, B, 0, stream>>>(src, dst, dinv, hbuf, agg, e);
    k_finalize<<<bfillh, B, 0, stream>>>(hbuf, dinv, b2, agg, nh);

    // final FC
    k_fc<<<bfill, B, 0, stream>>>(agg, Wfc, bfc, out, (int)n);
}